// CartTensorOut_54443005444463
// MI455X (gfx1250) — compile-verified
//
#include <hip/hip_runtime.h>
#include <hip/hip_bf16.h>

typedef float v2f __attribute__((ext_vector_type(2)));
typedef float v8f __attribute__((ext_vector_type(8)));

#define N_NODES   20000
#define N_GRAPH   256
#define OUT_ELEMS (N_GRAPH * 9)

// workspace layout (floats)
#define WS_WM2P 0        // [64][6][32]  = 12288
#define WS_BP   12288    // [6][32]      = 192
#define WS_C222 12480    // [5][5][5]    = 125
#define WS_TOTAL 12605

__device__ __forceinline__ v8f wmma4(v2f a, v2f b, v8f c) {
    // V_WMMA_F32_16X16X4_F32 : D = A(16x4,f32) x B(4x16,f32) + C(16x16,f32)
    return __builtin_amdgcn_wmma_f32_16x16x4_f32(false, a, false, b, (short)0, c,
                                                 false, false);
}

__constant__ float Q2C[5][9] = {
    {0.f, 0.70710678118654752f, 0.f, 0.70710678118654752f, 0.f, 0.f, 0.f, 0.f, 0.f},
    {0.f, 0.f, 0.f, 0.f, 0.f, 0.70710678118654752f, 0.f, 0.70710678118654752f, 0.f},
    {-0.40824829046386302f, 0.f, 0.f, 0.f, -0.40824829046386302f, 0.f, 0.f, 0.f,
     0.81649658092772603f},
    {0.f, 0.f, 0.70710678118654752f, 0.f, 0.f, 0.f, 0.70710678118654752f, 0.f, 0.f},
    {0.70710678118654752f, 0.f, 0.f, 0.f, -0.70710678118654752f, 0.f, 0.f, 0.f, 0.f}};

__global__ void zero_out_kernel(float* out, int n) {
    int i = blockIdx.x * blockDim.x + threadIdx.x;
    if (i < n) out[i] = 0.f;
}

// ---------------------------------------------------------------------------
// Prep: fold P vectors (and all scalar prefactors) into Wm2 / bm2, and build
// the C222 Clebsch-Gordan table (f64, matching the numpy construction).
// Effective paths: p0=k0, p1=k2, p2=k6 (scalar out) ; p3=k1+k5, p4=k4, p5=k8.
// ---------------------------------------------------------------------------
__global__ void prep_kernel(const float* __restrict__ Wm2, const float* __restrict__ bm2,
                            const float* __restrict__ P0, const float* __restrict__ P2,
                            float* __restrict__ ws) {
    const int tid = blockIdx.x * blockDim.x + threadIdx.x;
    const double SA = 0.018042195912175804;  // 1/sqrt(32*96)
    const double SB = 0.015625;              // 1/sqrt(32*128) = 1/64

    if (tid < 12288 + 192) {
        int idx, h;
        const float* src;
        if (tid < 12288) { h = tid / 192; idx = tid - h * 192; src = Wm2 + h * 9216; }
        else             { h = -1;        idx = tid - 12288;   src = bm2; }
        const int p = idx / 32, u = idx - p * 32;
        int k; const float* P; double sc;
        switch (p) {
            case 0: k = 0; P = P0;      sc = SA; break;
            case 1: k = 2; P = P0 + 32; sc = SA; break;
            case 2: k = 6; P = P0 + 64; sc = SA; break;
            case 3: k = 1; P = P2;      sc = SB; break;
            case 4: k = 4; P = P2 + 32; sc = SB; break;
            default: k = 8; P = P2 + 96; sc = SB; break;
        }
        double acc = 0.0;
        const float* row = src + k * 1024 + u * 32;
        for (int w = 0; w < 32; ++w) acc += (double)row[w] * (double)P[w];
        if (p == 3) {  // merge k=5 (identical CG path t = h0*h2_r)
            const float* row2 = src + 5 * 1024 + u * 32;
            const float* P5 = P2 + 64;
            for (int w = 0; w < 32; ++w) acc += (double)row2[w] * (double)P5[w];
        }
        ws[(tid < 12288 ? WS_WM2P : WS_BP) + (tid < 12288 ? tid : idx)] = (float)(acc * sc);
    } else if (tid == 12288 + 192) {
        // ---- build C222 in double precision ----
        double s2 = 0.7071067811865475244, s6 = 0.4082482904638630164;
        double Q2[5][3][3] = {};
        Q2[0][0][1] = Q2[0][1][0] = s2;
        Q2[1][1][2] = Q2[1][2][1] = s2;
        Q2[2][0][0] = Q2[2][1][1] = -s6; Q2[2][2][2] = 2.0 * s6;
        Q2[3][0][2] = Q2[3][2][0] = s2;
        Q2[4][0][0] = s2; Q2[4][1][1] = -s2;
        double Ct[5][5][5];
        double sumsq = 0.0;
        for (int i = 0; i < 5; ++i)
            for (int j = 0; j < 5; ++j) {
                double M[3][3];
                for (int a = 0; a < 3; ++a)
                    for (int d = 0; d < 3; ++d) {
                        double m = 0.0;
                        for (int b = 0; b < 3; ++b) m += Q2[i][a][b] * Q2[j][d][b];
                        M[a][d] = m;
                    }
                double S[3][3];
                for (int a = 0; a < 3; ++a)
                    for (int d = 0; d < 3; ++d) S[a][d] = 0.5 * (M[a][d] + M[d][a]);
                double tr = (S[0][0] + S[1][1] + S[2][2]) / 3.0;
                for (int e = 0; e < 3; ++e) S[e][e] -= tr;
                for (int kq = 0; kq < 5; ++kq) {
                    double c = 0.0;
                    for (int a = 0; a < 3; ++a)
                        for (int d = 0; d < 3; ++d) c += Q2[kq][a][d] * S[a][d];
                    Ct[i][j][kq] = c;
                    sumsq += c * c;
                }
            }
        double f = sqrt(5.0 / sumsq);
        for (int i = 0; i < 5; ++i)
            for (int j = 0; j < 5; ++j)
                for (int kq = 0; kq < 5; ++kq)
                    ws[WS_C222 + (i * 5 + j) * 5 + kq] = (float)(Ct[i][j][kq] * f);
    }
}

// ---------------------------------------------------------------------------
// Fused kernel: 64 threads = 2 waves, 16 nodes per wave.
//   Phase A (WMMA f32 16x16x4): hidden MLP + 9 spherical projections -> LDS.
//   Phase B (lane = u channel): wP = hidden @ Wm2P, CG contraction, butterfly
//   reduce over u, Cartesian basis change, rolled atomic segment-sum.
// ---------------------------------------------------------------------------
__global__ __launch_bounds__(64) void fused_kernel(
    const float* __restrict__ Xs, const float* __restrict__ Xsph,
    const int* __restrict__ batch, const float* __restrict__ W0e,
    const float* __restrict__ W1o, const float* __restrict__ W2e,
    const float* __restrict__ Wm1, const float* __restrict__ bm1,
    const float* __restrict__ ws, float* __restrict__ out) {
    __shared__ float sh_hidden[2][16 * 64];      // [wave][node][h]
    __shared__ float sh_h[2][16 * 9 * 32];       // [wave][node][comp][u]
    __shared__ float sh_C[125];

    const int tid = threadIdx.x;
    const int lane = tid & 31;
    const int wv = tid >> 5;
    if (tid < 125) sh_C[tid] = ws[WS_C222 + tid];

    const int lhalf = lane >> 4;   // 0: K lo pair, 1: K hi pair
    const int l15 = lane & 15;     // row (A) / column (B,C)
    const int nodeBase = (blockIdx.x * 2 + wv) * 16;
    int node = nodeBase + l15;
    if (node > N_NODES - 1) node = N_NODES - 1;
    const float* xrow = Xs + (size_t)node * 128;
    const float* srow = Xsph + (size_t)node * 480;

    // ---------- hidden = silu(Xs @ Wm1 + bm1) ----------
    v8f acc[4];
#pragma unroll
    for (int nt = 0; nt < 4; ++nt)
#pragma unroll
        for (int e = 0; e < 8; ++e) acc[nt][e] = 0.f;
#pragma unroll 4
    for (int kb = 0; kb < 32; ++kb) {
        const int k0 = kb * 4 + lhalf * 2;
        v2f a; a.x = xrow[k0]; a.y = xrow[k0 + 1];
#pragma unroll
        for (int nt = 0; nt < 4; ++nt) {
            const int col = l15 + nt * 16;
            v2f b; b.x = Wm1[k0 * 64 + col]; b.y = Wm1[(k0 + 1) * 64 + col];
            acc[nt] = wmma4(a, b, acc[nt]);
        }
    }
#pragma unroll
    for (int nt = 0; nt < 4; ++nt)
#pragma unroll
        for (int e = 0; e < 8; ++e) {
            const int row = e + lhalf * 8;
            const int col = l15 + nt * 16;
            float v = acc[nt][e] + bm1[col];
            sh_hidden[wv][row * 64 + col] = v / (1.f + expf(-v));  // silu
        }

    // ---------- spherical projections h = (x_l @ W_l) / sqrt(mul) ----------
    auto proj = [&](const float* abase, int astr, int kbN, const float* W,
                    float scale, int comp) {
        v8f pa[2];
#pragma unroll
        for (int nt = 0; nt < 2; ++nt)
#pragma unroll
            for (int e = 0; e < 8; ++e) pa[nt][e] = 0.f;
        for (int kb = 0; kb < kbN; ++kb) {
            const int k0 = kb * 4 + lhalf * 2;
            v2f a; a.x = abase[k0 * astr]; a.y = abase[(k0 + 1) * astr];
#pragma unroll
            for (int nt = 0; nt < 2; ++nt) {
                const int col = l15 + nt * 16;
                v2f b; b.x = W[k0 * 32 + col]; b.y = W[(k0 + 1) * 32 + col];
                pa[nt] = wmma4(a, b, pa[nt]);
            }
        }
#pragma unroll
        for (int nt = 0; nt < 2; ++nt)
#pragma unroll
            for (int e = 0; e < 8; ++e) {
                const int row = e + lhalf * 8;
                const int col = l15 + nt * 16;
                sh_h[wv][(row * 9 + comp) * 32 + col] = pa[nt][e] * scale;
            }
    };
    proj(srow, 1, 32, W0e, 0.088388347648318447f, 0);            // l=0, 1/sqrt(128)
#pragma unroll
    for (int m = 0; m < 3; ++m)
        proj(srow + 128 + m, 3, 16, W1o, 0.125f, 1 + m);         // l=1, 1/sqrt(64)
#pragma unroll
    for (int m = 0; m < 5; ++m)
        proj(srow + 320 + m, 5, 8, W2e, 0.17677669529663687f, 4 + m);  // l=2

    __syncthreads();

    // ---------- phase B: lane = u channel ----------
    const int u = lane;
    float pS0[16];
    float pS4[16][5];
#pragma unroll
    for (int n = 0; n < 16; ++n) {
        pS0[n] = 0.f;
#pragma unroll
        for (int r = 0; r < 5; ++r) pS4[n][r] = 0.f;
    }
    float wp[16];
    auto compute_wp = [&](int p) {
#pragma unroll
        for (int n = 0; n < 16; ++n) wp[n] = 0.f;
        for (int h = 0; h < 64; ++h) {
            const float wh = ws[WS_WM2P + h * 192 + p * 32 + u];
#pragma unroll
            for (int n = 0; n < 16; ++n) wp[n] += sh_hidden[wv][n * 64 + h] * wh;
        }
        const float bv = ws[WS_BP + p * 32 + u];
#pragma unroll
        for (int n = 0; n < 16; ++n) wp[n] += bv;
    };

    // p0: (0,0) C000 -> slot0, t = h0^2
    compute_wp(0);
#pragma unroll
    for (int n = 0; n < 16; ++n) {
        float c0 = sh_h[wv][(n * 9 + 0) * 32 + u];
        pS0[n] += wp[n] * (c0 * c0);
    }
    // p1: (1,1) C110 -> slot0, t = |h1|^2 / sqrt(3)
    compute_wp(1);
#pragma unroll
    for (int n = 0; n < 16; ++n) {
        float a0 = sh_h[wv][(n * 9 + 1) * 32 + u];
        float a1 = sh_h[wv][(n * 9 + 2) * 32 + u];
        float a2 = sh_h[wv][(n * 9 + 3) * 32 + u];
        pS0[n] += wp[n] * ((a0 * a0 + a1 * a1 + a2 * a2) * 0.57735026918962576f);
    }
    // p2: (2,2) C220 -> slot0, t = |h2|^2 / sqrt(5)
    compute_wp(2);
#pragma unroll
    for (int n = 0; n < 16; ++n) {
        float s = 0.f;
#pragma unroll
        for (int q = 0; q < 5; ++q) {
            float b = sh_h[wv][(n * 9 + 4 + q) * 32 + u];
            s += b * b;
        }
        pS0[n] += wp[n] * (s * 0.44721359549995794f);
    }
    // p3: k1 + k5 merged -> slots 4..8, t_r = h0 * h2_r
    compute_wp(3);
#pragma unroll
    for (int n = 0; n < 16; ++n) {
        float c0 = sh_h[wv][(n * 9 + 0) * 32 + u];
#pragma unroll
        for (int r = 0; r < 5; ++r)
            pS4[n][r] += wp[n] * c0 * sh_h[wv][(n * 9 + 4 + r) * 32 + u];
    }
    // p4: (1,1) C112 -> slots 4..8, t_r = Q2[r] : h1 h1
    compute_wp(4);
    {
        const float S2 = 0.70710678118654752f, S6 = 0.40824829046386302f;
#pragma unroll
        for (int n = 0; n < 16; ++n) {
            float a0 = sh_h[wv][(n * 9 + 1) * 32 + u];
            float a1 = sh_h[wv][(n * 9 + 2) * 32 + u];
            float a2 = sh_h[wv][(n * 9 + 3) * 32 + u];
            pS4[n][0] += wp[n] * (2.f * S2 * a0 * a1);
            pS4[n][1] += wp[n] * (2.f * S2 * a1 * a2);
            pS4[n][2] += wp[n] * (S6 * (2.f * a2 * a2 - a0 * a0 - a1 * a1));
            pS4[n][3] += wp[n] * (2.f * S2 * a0 * a2);
            pS4[n][4] += wp[n] * (S2 * (a0 * a0 - a1 * a1));
        }
    }
    // p5: (2,2) C222 -> slots 4..8
    compute_wp(5);
#pragma unroll
    for (int n = 0; n < 16; ++n) {
        float b[5];
#pragma unroll
        for (int q = 0; q < 5; ++q) b[q] = sh_h[wv][(n * 9 + 4 + q) * 32 + u];
        float t[5] = {0.f, 0.f, 0.f, 0.f, 0.f};
#pragma unroll
        for (int i = 0; i < 5; ++i)
#pragma unroll
            for (int j = 0; j < 5; ++j) {
                float bij = b[i] * b[j];
#pragma unroll
                for (int r = 0; r < 5; ++r) t[r] += sh_C[(i * 5 + j) * 5 + r] * bij;
            }
#pragma unroll
        for (int r = 0; r < 5; ++r) pS4[n][r] += wp[n] * t[r];
    }

    // ---------- reduce over u, Cartesian map, rolled segment-sum ----------
#pragma unroll
    for (int n = 0; n < 16; ++n) {
        float v0 = pS0[n];
        float v1 = pS4[n][0], v2 = pS4[n][1], v3 = pS4[n][2], v4 = pS4[n][3],
              v5 = pS4[n][4];
        for (int off = 16; off > 0; off >>= 1) {
            v0 += __shfl_xor(v0, off, 32);
            v1 += __shfl_xor(v1, off, 32);
            v2 += __shfl_xor(v2, off, 32);
            v3 += __shfl_xor(v3, off, 32);
            v4 += __shfl_xor(v4, off, 32);
            v5 += __shfl_xor(v5, off, 32);
        }
        const int nd = nodeBase + n;
        if (lane < 9 && nd < N_NODES) {
            const int ai = lane / 3, bi = lane - ai * 3;
            float c = (ai == bi) ? v0 * 0.57735026918962576f : 0.f;
            c += v1 * Q2C[0][lane] + v2 * Q2C[1][lane] + v3 * Q2C[2][lane] +
                 v4 * Q2C[3][lane] + v5 * Q2C[4][lane];
            const int g = batch[nd];
            // jnp.roll(+1, axis=1&2): cart (a,b) lands at ((a+1)%3, (b+1)%3)
            atomicAdd(out + g * 9 + ((ai + 1) % 3) * 3 + ((bi + 1) % 3), c);
        }
    }
}

extern "C" void kernel_launch(void* const* d_in, const int* in_sizes, int n_in,
                              void* d_out, int out_size, void* d_ws, size_t ws_size,
                              hipStream_t stream) {
    const float* x_scalar = (const float*)d_in[0];
    const float* x_sph    = (const float*)d_in[1];
    const int*   batch    = (const int*)d_in[2];
    const float* W0e      = (const float*)d_in[3];
    const float* W1o      = (const float*)d_in[4];
    const float* W2e      = (const float*)d_in[5];
    const float* Wm1      = (const float*)d_in[6];
    const float* bm1      = (const float*)d_in[7];
    const float* Wm2      = (const float*)d_in[8];
    const float* bm2      = (const float*)d_in[9];
    const float* P0       = (const float*)d_in[10];
    const float* P2       = (const float*)d_in[12];
    float* out = (float*)d_out;
    float* ws  = (float*)d_ws;

    zero_out_kernel<<<dim3((OUT_ELEMS + 255) / 256), dim3(256), 0, stream>>>(out,
                                                                             OUT_ELEMS);
    prep_kernel<<<dim3(49), dim3(256), 0, stream>>>(Wm2, bm2, P0, P2, ws);
    fused_kernel<<<dim3(N_NODES / 32), dim3(64), 0, stream>>>(
        x_scalar, x_sph, batch, W0e, W1o, W2e, Wm1, bm1, ws, out);
}